// GptOssTopKRouter_75041668596190
// MI455X (gfx1250) — compile-verified
//
#include <hip/hip_runtime.h>

typedef float v2f  __attribute__((ext_vector_type(2)));
typedef float v8f  __attribute__((ext_vector_type(8)));
typedef unsigned int u32x4 __attribute__((ext_vector_type(4)));
typedef int i32x4 __attribute__((ext_vector_type(4)));
typedef int i32x8 __attribute__((ext_vector_type(8)));

#define HDIM 4096
#define NEXP 8
#define BK   32                 // K-chunk (power-of-two => legal TDM pad interval)
#define RSTR 36                 // padded row stride in floats (bank-conflict-free)
#define TPW  16                 // tokens per wave (WMMA M)
#define WPB  8                  // waves per block
#define TPB  (TPW * WPB)        // 128 tokens per block
#define NCH  (HDIM / BK)        // 128 K-chunks

#if defined(__has_builtin)
#  if __has_builtin(__builtin_amdgcn_tensor_load_to_lds) && __has_builtin(__builtin_amdgcn_s_wait_tensorcnt)
#    define USE_TDM 1
#  else
#    define USE_TDM 0
#  endif
#else
#  define USE_TDM 0
#endif

__device__ __forceinline__ unsigned lds_byte_offset(const void* p) {
    return (unsigned)(unsigned long long)(__attribute__((address_space(3))) const void*)p;
}

__global__ __launch_bounds__(256) void GptOssTopKRouter_kernel(
    const float* __restrict__ hs,     // [T, H]
    const float* __restrict__ Wg,     // [H, E]
    const float* __restrict__ bg,     // [E]
    float* __restrict__ scores,       // [T, E]
    float* __restrict__ idx_out,      // [T, 2] (index values stored as float)
    int T)
{
    __shared__ float tiles[2][WPB][TPW * RSTR];   // 36,864 B, double-buffered per-wave tiles

    const int tid   = threadIdx.x;
    const int wave  = __builtin_amdgcn_readfirstlane(tid >> 5);   // wave-uniform in SGPR
    const int lane  = tid & 31;
    const int half  = lane >> 4;       // 0: lanes 0-15, 1: lanes 16-31
    const int laneM = lane & 15;

    const int tile0 = blockIdx.x * TPB + wave * TPW;
    if (tile0 >= T) return;            // whole wave exits uniformly; no barriers below

    // B-fragment column select + mask: keeps loads in-bounds and EXEC uniform at WMMA.
    const int   ecol  = (laneM < NEXP) ? laneM : 0;
    const float emask = (laneM < NEXP) ? 1.0f : 0.0f;

    const unsigned long long gbase =
        (unsigned long long)(size_t)(hs + (size_t)tile0 * HDIM);   // tile row 0, byte addr

#if USE_TDM
    // Loop-invariant D# group 1: 4-byte elems; LDS pad every 32 DWORDs by 4 DWORDs (=> stride 36);
    // tensor = [T rows x HDIM cols], tile = [16 rows x 32 cols], row stride HDIM.
    i32x8 g1;
    g1[0] = (int)((2u << 16) | (1u << 20) | (4u << 22) | (3u << 25));
    g1[1] = (int)(((unsigned)HDIM & 0xFFFFu) << 16);                       // tensor_dim0 lo16
    g1[2] = (int)(((unsigned)HDIM >> 16) | (((unsigned)T & 0xFFFFu) << 16)); // dim0 hi | dim1 lo
    g1[3] = (int)((((unsigned)T >> 16) & 0xFFFFu) | ((unsigned)BK << 16)); // dim1 hi | tile_dim0
    g1[4] = TPW;                                                           // tile_dim1=16, tile_dim2=0
    g1[5] = HDIM;                                                          // tensor_dim0_stride lo32
    g1[6] = 0;
    g1[7] = 0;
    const i32x4 gz = {0, 0, 0, 0};
#endif

    auto issue_tile = [&](int kb, int buf) {
#if USE_TDM
        const unsigned long long ga = gbase + (unsigned long long)kb * 4u;
        u32x4 g0;
        g0[0] = 1u;                                            // count=1, user descriptor
        g0[1] = lds_byte_offset(&tiles[buf][wave][0]);         // LDS byte address
        g0[2] = (unsigned)(ga & 0xFFFFFFFFu);                  // global addr [31:0]
        g0[3] = (unsigned)((ga >> 32) & 0x01FFFFFFu) | (2u << 30);  // addr [56:32] | type=2
#  if __clang_major__ >= 23
        i32x8 gz8 = {0, 0, 0, 0, 0, 0, 0, 0};
        __builtin_amdgcn_tensor_load_to_lds(g0, g1, gz, gz, gz8, 0);
#  else
        __builtin_amdgcn_tensor_load_to_lds(g0, g1, gz, gz, 0);
#  endif
#else
        // Fallback: coalesced float4 copy (8-lane group covers one 128B row segment).
        float* A = &tiles[buf][wave][0];
        #pragma unroll
        for (int p = 0; p < 4; ++p) {
            int row = p * 4 + (lane >> 3);
            int q   = (lane & 7) * 4;
            int tok = tile0 + row;
            float4 v = make_float4(0.f, 0.f, 0.f, 0.f);
            if (tok < T) v = *(const float4*)&hs[(size_t)tok * HDIM + kb + q];
            *(float4*)&A[row * RSTR + q] = v;
        }
#endif
    };

    v8f accA = {};   // even K-subtiles
    v8f accB = {};   // odd  K-subtiles

    issue_tile(0, 0);

    for (int c = 0; c < NCH; ++c) {
        const int cur = c & 1;
        if (c + 1 < NCH) {
            issue_tile((c + 1) * BK, cur ^ 1);     // prefetch next chunk into other buffer
#if USE_TDM
            __builtin_amdgcn_s_wait_tensorcnt(1);  // oldest (current) tile complete
#endif
        } else {
#if USE_TDM
            __builtin_amdgcn_s_wait_tensorcnt(0);
#endif
        }

        const float* A  = &tiles[cur][wave][0];
        const float* Wk = &Wg[(size_t)c * BK * NEXP];   // W chunk: L2/WGP$-resident (128 KB total)

        #pragma unroll
        for (int k4 = 0; k4 < BK / 4; ++k4) {
            const int kk = k4 * 4 + half * 2;      // A/B VGPR0/1 hold K = 2h, 2h+1
            v2f a, b;
            a.x = A[laneM * RSTR + kk];
            a.y = A[laneM * RSTR + kk + 1];
            b.x = Wk[(size_t)(kk)     * NEXP + ecol] * emask;  // zero for padded cols 8..15
            b.y = Wk[(size_t)(kk + 1) * NEXP + ecol] * emask;
            if (k4 & 1)
                accB = __builtin_amdgcn_wmma_f32_16x16x4_f32(
                           false, a, false, b, (short)0, accB, false, false);
            else
                accA = __builtin_amdgcn_wmma_f32_16x16x4_f32(
                           false, a, false, b, (short)0, accA, false, false);
        }
    }

    v8f acc = accA + accB;

    const float be  = (laneM < NEXP) ? bg[laneM] : 0.0f;
    const float NEG = -__builtin_inff();

    // C/D layout: acc[r] on lane = logits[token = tile0 + half*8 + r][expert = laneM]
    #pragma unroll
    for (int r = 0; r < 8; ++r) {
        const int  token = tile0 + half * 8 + r;
        const bool valid = (laneM < NEXP);

        float v1 = valid ? (acc[r] + be) : NEG;
        int   i1 = laneM;
        float v2 = NEG;
        int   i2 = 0x7fffffff;

        // top-2 across the 8-lane expert group (low index wins ties, like lax.top_k)
        #pragma unroll
        for (int off = 1; off < 8; off <<= 1) {
            float ov1 = __shfl_xor(v1, off, 8);
            int   oi1 = __shfl_xor(i1, off, 8);
            float ov2 = __shfl_xor(v2, off, 8);
            int   oi2 = __shfl_xor(i2, off, 8);
            bool oFirst = (ov1 > v1) || (ov1 == v1 && oi1 < i1);
            float n1, n2; int ni1, ni2;
            if (oFirst) {
                n1 = ov1; ni1 = oi1;
                bool s = (v1 > ov2) || (v1 == ov2 && i1 < oi2);
                n2 = s ? v1 : ov2; ni2 = s ? i1 : oi2;
            } else {
                n1 = v1; ni1 = i1;
                bool s = (ov1 > v2) || (ov1 == v2 && oi1 < i2);
                n2 = s ? ov1 : v2; ni2 = s ? oi1 : i2;
            }
            v1 = n1; i1 = ni1; v2 = n2; i2 = ni2;
        }

        // stable 2-way softmax (v1 >= v2)
        float ex = expf(v2 - v1);
        float p1 = 1.0f / (1.0f + ex);
        float p2 = ex * p1;

        if (valid && token < T) {
            float s = (laneM == i1) ? p1 : ((laneM == i2) ? p2 : 0.0f);
            scores[(size_t)token * NEXP + laneM] = s;
        }
        if (laneM == 0 && token < T) {
            idx_out[(size_t)token * 2 + 0] = (float)i1;
            idx_out[(size_t)token * 2 + 1] = (float)i2;
        }
    }
}

extern "C" void kernel_launch(void* const* d_in, const int* in_sizes, int n_in,
                              void* d_out, int out_size, void* d_ws, size_t ws_size,
                              hipStream_t stream) {
    const float* hs = (const float*)d_in[0];
    const float* Wg = (const float*)d_in[1];
    const float* bg = (const float*)d_in[2];
    const int T = in_sizes[0] / HDIM;            // B*S tokens
    float* scores  = (float*)d_out;              // [T, 8]
    float* idx_out = scores + (size_t)T * NEXP;  // [T, 2] as float values
    const int blocks = (T + TPB - 1) / TPB;
    hipLaunchKernelGGL(GptOssTopKRouter_kernel, dim3(blocks), dim3(256), 0, stream,
                       hs, Wg, bg, scores, idx_out, T);
}